// CARAFEPack_29781303231117
// MI455X (gfx1250) — compile-verified
//
#include <hip/hip_runtime.h>
#include <math.h>

// CARAFE upsample, fp32 end-to-end, WMMA F32_16X16X4_F32 for both convs.
// Wave32 layouts (CDNA5 ISA 7.12.2):
//   A 16x4 f32 : lane L -> M = L%16, VGPR v -> K = 2*(L/16)+v   (v2f per lane)
//   B 4x16 f32 : lane L -> N = L%16, VGPR v -> K = 2*(L/16)+v   (v2f per lane)
//   C/D 16x16  : lane L -> N = L%16, VGPR v -> M = v + 8*(L/16) (v8f per lane)

typedef __attribute__((ext_vector_type(2))) float v2f;
typedef __attribute__((ext_vector_type(8))) float v8f;

#define HW 64
#define PIX 4096              // 64*64
#define CIN 256
#define CCOMP 64
#define MENC 100
#define MPAD 112              // 7 tiles of 16
#define KENC 576              // 64*9
#define NPIX_TOTAL 16384      // 4 * 4096

// ---------------------------------------------------------------------------
// Kernel 1: 1x1 conv (channel compressor).  GEMM: M=64, K=256, N=16384.
// One wave -> 16 pixels x all 64 out channels (4 C tiles).  comp layout [n][64][4096].
// ---------------------------------------------------------------------------
__global__ __launch_bounds__(256) void carafe_compress(
    const float* __restrict__ x, const float* __restrict__ w_comp,
    const float* __restrict__ b_comp, float* __restrict__ comp) {
  const int wave = threadIdx.x >> 5;
  const int lane = threadIdx.x & 31;
  const int row  = lane & 15;
  const int half = lane >> 4;

  const int tile = blockIdx.x * 8 + wave;     // 0..1023
  const int gp   = tile << 4;                 // global pixel base
  const int n    = gp >> 12;
  const int pin  = gp & 4095;

  v8f acc[4];
#pragma unroll
  for (int t = 0; t < 4; ++t)
#pragma unroll
    for (int v = 0; v < 8; ++v)
      acc[t][v] = b_comp[t * 16 + v + 8 * half];

  const float* xb = x + (size_t)n * CIN * PIX + pin + row;   // B: pixel = pin+row

  for (int k0 = 0; k0 < CIN; k0 += 4) {
    const int kb = k0 + 2 * half;
    v2f b;
    b[0] = xb[(size_t)(kb + 0) * PIX];
    b[1] = xb[(size_t)(kb + 1) * PIX];
#pragma unroll
    for (int t = 0; t < 4; ++t) {
      v2f a = *(const v2f*)(w_comp + (t * 16 + row) * CIN + kb);
      acc[t] = __builtin_amdgcn_wmma_f32_16x16x4_f32(
          false, a, false, b, (short)0, acc[t], false, false);
    }
  }

  float* cb = comp + (size_t)n * CCOMP * PIX + pin + row;
#pragma unroll
  for (int t = 0; t < 4; ++t)
#pragma unroll
    for (int v = 0; v < 8; ++v)
      cb[(size_t)(t * 16 + v + 8 * half) * PIX] = acc[t][v];
}

// ---------------------------------------------------------------------------
// Kernel 2: 3x3 conv (content encoder) + pixel_shuffle softmax, fused.
// GEMM via im2col: M=100 (pad 112), K=576, N=16384.  One wave -> 16 pixels,
// all 7 M tiles in registers; softmax over 25 taps per (pixel, i, j) via LDS.
// mask layout: [n*4096 + p][ij][k]  (100 contiguous floats per pixel).
// ---------------------------------------------------------------------------
__global__ __launch_bounds__(256) void carafe_encode_softmax(
    const float* __restrict__ comp, const float* __restrict__ w_enc,
    const float* __restrict__ b_enc, float* __restrict__ mask) {
  __shared__ float sm[8][16 * 113];

  const int wave = threadIdx.x >> 5;
  const int lane = threadIdx.x & 31;
  const int row  = lane & 15;
  const int half = lane >> 4;

  const int tile = blockIdx.x * 8 + wave;     // 0..1023
  const int gp   = tile << 4;
  const int n    = gp >> 12;
  const int pin  = gp & 4095;

  const int p  = pin + row;                   // this lane's B pixel
  const int py = p >> 6;
  const int px = p & 63;

  v8f acc[7];
#pragma unroll
  for (int t = 0; t < 7; ++t)
#pragma unroll
    for (int v = 0; v < 8; ++v) {
      const int m = t * 16 + v + 8 * half;
      acc[t][v] = (m < MENC) ? b_enc[m] : 0.0f;
    }

  const float* cbase = comp + (size_t)n * CCOMP * PIX;

  for (int k0 = 0; k0 < KENC; k0 += 4) {
    const int kb = k0 + 2 * half;
    v2f b;
#pragma unroll
    for (int v = 0; v < 2; ++v) {
      const int k  = kb + v;
      const int cc = k / 9;
      const int r  = k - cc * 9;
      const int y  = py + (r / 3) - 1;
      const int xx = px + (r % 3) - 1;
      const bool ok = ((unsigned)y < HW) & ((unsigned)xx < HW);
      const int yc = min(max(y, 0), HW - 1);
      const int xc = min(max(xx, 0), HW - 1);
      const float val = cbase[(size_t)cc * PIX + yc * HW + xc];  // always in-range addr
      b[v] = ok ? val : 0.0f;                                    // v_cndmask, EXEC stays full
    }
#pragma unroll
    for (int t = 0; t < 7; ++t) {
      const int m  = t * 16 + row;
      const int mc = min(m, MENC - 1);
      v2f a = *(const v2f*)(w_enc + (size_t)mc * KENC + kb);
      if (m >= MENC) { a[0] = 0.0f; a[1] = 0.0f; }
      acc[t] = __builtin_amdgcn_wmma_f32_16x16x4_f32(
          false, a, false, b, (short)0, acc[t], false, false);
    }
  }

  // Stage D tiles to LDS: sm[wave][pixel*113 + channel]
  float* s = sm[wave];
#pragma unroll
  for (int t = 0; t < 7; ++t)
#pragma unroll
    for (int v = 0; v < 8; ++v)
      s[row * 113 + t * 16 + v + 8 * half] = acc[t][v];

  __syncthreads();

  // Softmax: 64 groups (16 pixels x 4 sub-positions) -> 2 per lane.
#pragma unroll
  for (int u = 0; u < 2; ++u) {
    const int g   = lane * 2 + u;
    const int pix = g >> 2;
    const int ij  = g & 3;
    const float* base = s + pix * 113;
    float r[25];
    float mx = -3.4e38f;
#pragma unroll
    for (int k = 0; k < 25; ++k) { r[k] = base[k * 4 + ij]; mx = fmaxf(mx, r[k]); }
    float sum = 0.0f;
#pragma unroll
    for (int k = 0; k < 25; ++k) { r[k] = __expf(r[k] - mx); sum += r[k]; }
    const float inv = 1.0f / sum;
    float* mout = mask + (size_t)(gp + pix) * 100 + ij * 25;
#pragma unroll
    for (int k = 0; k < 25; ++k) mout[k] = r[k] * inv;
  }
}

// ---------------------------------------------------------------------------
// Kernel 3: CARAFE reassembly.  out[n,c,2h+i,2w+j] = sum_k x5x5[k] * mask[p][ij][k]
// Block = 64 threads (one w-row), one (n, h, group of 8 channels).
// Mask row cached in LDS (stride 101 -> bank-conflict-free), 8x reuse.
// ---------------------------------------------------------------------------
__global__ __launch_bounds__(64) void carafe_reassemble(
    const float* __restrict__ x, const float* __restrict__ mask,
    float* __restrict__ out) {
  __shared__ float sm[64 * 101];

  const int t  = threadIdx.x;                 // w coordinate
  const int b  = blockIdx.x;
  const int cg = b & 31;                      // channel group (8 channels)
  const int h  = (b >> 5) & 63;
  const int n  = b >> 11;

  // Load this row's mask into LDS: 100 floats/pixel as 25 float4s.
  const int pg = (n << 12) + (h << 6) + t;
  const float4* mp4 = (const float4*)(mask + (size_t)pg * 100);
  float* srow = sm + t * 101;
#pragma unroll
  for (int q = 0; q < 25; ++q) {
    const float4 m4 = mp4[q];
    srow[q * 4 + 0] = m4.x; srow[q * 4 + 1] = m4.y;
    srow[q * 4 + 2] = m4.z; srow[q * 4 + 3] = m4.w;
  }
  __syncthreads();

  const float* ms = sm + t * 101;             // [ij*25 + k]

  for (int ci = 0; ci < 8; ++ci) {
    const int c = cg * 8 + ci;
    const float* xb = x + ((size_t)(n * 256 + c)) * PIX;

    float xv[25];
#pragma unroll
    for (int di = 0; di < 5; ++di) {
      const int y  = h + di - 2;
      const int yc = min(max(y, 0), HW - 1);
#pragma unroll
      for (int dj = 0; dj < 5; ++dj) {
        const int xx = t + dj - 2;
        const int xc = min(max(xx, 0), HW - 1);
        const bool ok = ((unsigned)y < HW) & ((unsigned)xx < HW);
        const float val = xb[yc * HW + xc];
        xv[di * 5 + dj] = ok ? val : 0.0f;
      }
    }

    float a0 = 0.f, a1 = 0.f, a2 = 0.f, a3 = 0.f;
#pragma unroll
    for (int k = 0; k < 25; ++k) {
      const float pch = xv[k];
      a0 += pch * ms[k];
      a1 += pch * ms[25 + k];
      a2 += pch * ms[50 + k];
      a3 += pch * ms[75 + k];
    }

    float* ob = out + (((size_t)(n * 256 + c) * 128) + 2 * h) * 128 + 2 * t;
    float2 r0; r0.x = a0; r0.y = a1;
    float2 r1; r1.x = a2; r1.y = a3;
    *(float2*)(ob)       = r0;   // row 2h,   cols 2t, 2t+1
    *(float2*)(ob + 128) = r1;   // row 2h+1, cols 2t, 2t+1
  }
}

// ---------------------------------------------------------------------------
extern "C" void kernel_launch(void* const* d_in, const int* in_sizes, int n_in,
                              void* d_out, int out_size, void* d_ws, size_t ws_size,
                              hipStream_t stream) {
  (void)in_sizes; (void)n_in; (void)out_size; (void)ws_size;
  const float* x      = (const float*)d_in[0];
  const float* w_comp = (const float*)d_in[1];
  const float* b_comp = (const float*)d_in[2];
  const float* w_enc  = (const float*)d_in[3];
  const float* b_enc  = (const float*)d_in[4];
  float* out = (float*)d_out;

  float* comp = (float*)d_ws;                      // 4*64*4096 f32 = 4.19 MB
  float* mask = comp + (size_t)4 * CCOMP * PIX;    // 16384*100 f32 = 6.55 MB

  // 1024 pixel tiles, 8 waves (tiles) per 256-thread block.
  carafe_compress<<<128, 256, 0, stream>>>(x, w_comp, b_comp, comp);
  carafe_encode_softmax<<<128, 256, 0, stream>>>(comp, w_enc, b_enc, mask);
  // 4 n * 64 h * 32 channel-groups = 8192 blocks of 64 threads.
  carafe_reassemble<<<8192, 64, 0, stream>>>(x, mask, out);
}